// HiPAttention_41712722379425
// MI455X (gfx1250) — compile-verified
//
#include <hip/hip_runtime.h>
#include <math.h>

#define BB 2
#define SS 1024
#define DD 1024
#define HH 16
#define DH 64

typedef float v2f __attribute__((ext_vector_type(2)));
typedef float v8f __attribute__((ext_vector_type(8)));
typedef unsigned int v4u __attribute__((ext_vector_type(4)));
typedef int v4i __attribute__((ext_vector_type(4)));
typedef int v8i __attribute__((ext_vector_type(8)));

// ---------------------------------------------------------------------------
// TDM: issue a 2D tensor_load_to_lds (6-arg toolchain form).
//   tile_d0 elems per row (fp32), tile_d1 rows, row stride in elems.
//   LDS padding: +4 dwords after every 32 dwords -> padded row stride 36.
// D# bitfield packing per CDNA5 ISA 8.3/8.4.
// ---------------------------------------------------------------------------
__device__ __forceinline__ void tdm_load_2d(unsigned lds_addr,
                                            const float* gptr, int tile_d0,
                                            int tile_d1, int row_stride) {
  const unsigned long long ga = (unsigned long long)(uintptr_t)gptr;
  v4u g0;
  g0[0] = 1u;                                  // count=1 (valid), user mode
  g0[1] = lds_addr;                            // lds_addr [63:32]
  g0[2] = (unsigned)(ga & 0xffffffffu);        // global_addr [95:64]
  g0[3] = (unsigned)((ga >> 32) & 0x01ffffffu) | (2u << 30);  // addr hi | type=2
  v8i g1;
  // data_size=2 (4B) <<16 | pad_enable<<20 | pad_interval=4 (32dw)<<22 |
  // pad_amount=3 (4dw)<<25 ; workgroup_mask=0 (not in cluster)
  g1[0] = (2 << 16) | (1 << 20) | (4 << 22) | (3 << 25);
  g1[1] = (int)((unsigned)tile_d0 << 16);      // tensor_dim0[15:0] @ bits79:48
  g1[2] = (int)((unsigned)tile_d1 << 16);      // tensor_dim1[15:0] @ bits95:80
  g1[3] = (int)((unsigned)tile_d0 << 16);      // tile_dim0 @ bits127:112
  g1[4] = tile_d1;                             // tile_dim1 @ bits143:128
  g1[5] = row_stride;                          // tensor_dim0_stride lo
  g1[6] = 0;
  g1[7] = 0;
  const v4i z4 = {0, 0, 0, 0};
  const v8i z8 = {0, 0, 0, 0, 0, 0, 0, 0};
  __builtin_amdgcn_tensor_load_to_lds(g0, g1, z4, z4, z8, 0);
}

// ---------------------------------------------------------------------------
// GEMM: out = A[M,K] @ W[N,K]^T + bias[N]
//   MODE 0: A is plain row-major (x [B*S, D]); store remapped to [B,H,S,DH]
//   MODE 1: A gathered from ctx [B,H,S,DH] as [B*S, D]; store plain [B*S, D]
// M = 2048, N = 1024, K = 1024. V_WMMA_F32_16X16X4_F32 (exact fp32), TDM
// double-buffered LDS staging.
// ---------------------------------------------------------------------------
template <int MODE>
__global__ __launch_bounds__(256) void gemm_wmma_f32(
    const float* __restrict__ A, const float* __restrict__ W,
    const float* __restrict__ bias, float* __restrict__ out) {
  constexpr int BM = 64, BN = 128, BK = 32;
  constexpr int LDT = BK + 4;  // padded LDS stride (36 floats)
  __shared__ __align__(16) float sA[2][BM][LDT];
  __shared__ __align__(16) float sW[2][BN][LDT];

  const int tid = threadIdx.x;
  const int lane = tid & 31;
  const int wv = tid >> 5;  // 8 waves
  const int wm = wv >> 2;   // 0..1 (M dir)
  const int wn = wv & 3;    // 0..3 (N dir)
  const int bm0 = blockIdx.y * BM;
  const int bn0 = blockIdx.x * BN;
  const bool issuer = (wv == 0);

  v8f acc[2][2];
#pragma unroll
  for (int mt = 0; mt < 2; ++mt)
#pragma unroll
    for (int nt = 0; nt < 2; ++nt)
      acc[mt][nt] = (v8f){0.f, 0.f, 0.f, 0.f, 0.f, 0.f, 0.f, 0.f};

  // A-tile global base for a given k0 (both modes are plain 2D tiles:
  // MODE 1 keeps each 32-wide K slice inside one head, row stride 64).
  auto a_base = [&](int k0) -> const float* {
    if (MODE == 0) {
      return A + (size_t)bm0 * DD + k0;
    } else {
      const int b = bm0 >> 10, s0 = bm0 & 1023;
      const int h = k0 >> 6, dh0 = k0 & 63;
      return A + (((size_t)b * HH + h) * SS + s0) * DH + dh0;
    }
  };
  constexpr int A_STRIDE = (MODE == 0) ? DD : DH;

  // prologue: stage k0 = 0 into buffer 0
  if (issuer) {
    tdm_load_2d((unsigned)(uintptr_t)&sA[0][0][0], a_base(0), BK, BM, A_STRIDE);
    tdm_load_2d((unsigned)(uintptr_t)&sW[0][0][0], W + (size_t)bn0 * DD + 0,
                BK, BN, DD);
    __builtin_amdgcn_s_wait_tensorcnt(0);
  }
  __syncthreads();

  for (int k0 = 0; k0 < DD; k0 += BK) {
    const int p = (k0 >> 5) & 1;
    // issue next slice into the other buffer (overlaps with WMMA work)
    if (issuer && (k0 + BK) < DD) {
      const int kn = k0 + BK;
      tdm_load_2d((unsigned)(uintptr_t)&sA[p ^ 1][0][0], a_base(kn), BK, BM,
                  A_STRIDE);
      tdm_load_2d((unsigned)(uintptr_t)&sW[p ^ 1][0][0],
                  W + (size_t)bn0 * DD + kn, BK, BN, DD);
    }

    const int koff = (lane >> 4) * 2;  // lanes 0-15 -> K{0,1}, 16-31 -> K{2,3}
    const int rl = lane & 15;
#pragma unroll
    for (int ks = 0; ks < BK; ks += 4) {
      v2f af[2], bf[2];
#pragma unroll
      for (int mt = 0; mt < 2; ++mt)
        af[mt] = *reinterpret_cast<const v2f*>(
            &sA[p][wm * 32 + mt * 16 + rl][ks + koff]);
#pragma unroll
      for (int nt = 0; nt < 2; ++nt)
        bf[nt] = *reinterpret_cast<const v2f*>(
            &sW[p][wn * 32 + nt * 16 + rl][ks + koff]);
#pragma unroll
      for (int mt = 0; mt < 2; ++mt)
#pragma unroll
        for (int nt = 0; nt < 2; ++nt)
          acc[mt][nt] = __builtin_amdgcn_wmma_f32_16x16x4_f32(
              false, af[mt], false, bf[nt], (short)0, acc[mt][nt], false,
              false);
    }
    // wave0 waits for the in-flight TDM, then all waves sync
    if (issuer) __builtin_amdgcn_s_wait_tensorcnt(0);
    __syncthreads();
  }

  // epilogue: C/D 16x16 layout -> VGPR g: lanes0-15 M=g, lanes16-31 M=g+8
  const int rl = lane & 15;
  const int mh = (lane >> 4) * 8;
#pragma unroll
  for (int mt = 0; mt < 2; ++mt)
#pragma unroll
    for (int nt = 0; nt < 2; ++nt) {
      const int colg = bn0 + wn * 32 + nt * 16 + rl;
      const float bv = bias[colg];
#pragma unroll
      for (int g = 0; g < 8; ++g) {
        const int rowg = bm0 + wm * 32 + mt * 16 + g + mh;
        const float v = acc[mt][nt][g] + bv;
        if (MODE == 0) {
          const int b = rowg >> 10, s = rowg & 1023;
          const int h = colg >> 6, dh = colg & 63;
          out[(((size_t)b * HH + h) * SS + s) * DH + dh] = v;
        } else {
          out[(size_t)rowg * DD + colg] = v;
        }
      }
    }
}

// ---------------------------------------------------------------------------
// Hierarchical pruning + sparse attention.
// One block per query i (1024 threads = 32 waves); wave w handles (b,h).
// Cross-(b,h) coupling: max_valid is the global max over all 32 waves.
// ---------------------------------------------------------------------------
__device__ __forceinline__ float dot64(const float* __restrict__ qlds,
                                       const float* __restrict__ kr) {
  float acc = 0.f;
#pragma unroll
  for (int j = 0; j < DH; j += 4) {
    const float4 kv = *reinterpret_cast<const float4*>(kr + j);
    const float4 qv = *reinterpret_cast<const float4*>(qlds + j);
    acc = fmaf(qv.x, kv.x, acc);
    acc = fmaf(qv.y, kv.y, acc);
    acc = fmaf(qv.z, kv.z, acc);
    acc = fmaf(qv.w, kv.w, acc);
  }
  return acc;
}

template <int LV>
__device__ __forceinline__ void prune_level(int i, int w, int lane,
                                            int& cur_len, const float* kbh,
                                            const float* qlds,
                                            int (*s_idx)[256], int* s_cnt) {
  constexpr int C = 32 >> LV;  // 32,16,8,4,2
  constexpr int NC = (LV == 0) ? 32 : 16;
  constexpr int NEWL = C * 8;  // 256,128,64,32,16
  constexpr int NR = (NEWL + 31) / 32;
  constexpr int SH = 5 - LV;  // log2(C)

  // chunk-representative scores (one chunk per lane)
  float sc = -1e30f;
  if (lane < NC) {
    const int repPos = lane * C + (C >> 1);
    int repVal;
    if (LV == 0)
      repVal = (repPos <= i) ? repPos : -1;
    else
      repVal = s_idx[w][repPos];
    const int rep = repVal < 0 ? 0 : repVal;  // -1 -> 0, as in torch
    const float acc = dot64(qlds, kbh + (size_t)rep * DH);
    if ((lane * C) < cur_len) sc = acc;  // phantom chunks -> -inf
  }

  // top-8: iterative wave argmax; ties -> lower index (matches lax.top_k)
  unsigned long long pack = 0ull;
  bool taken = false;
#pragma unroll
  for (int t = 0; t < 8; ++t) {
    float bs = taken ? -3.0e38f : sc;
    int bi = taken ? (64 + lane) : lane;
#pragma unroll
    for (int d = 16; d > 0; d >>= 1) {
      const float os = __shfl_xor(bs, d, 32);
      const int oi = __shfl_xor(bi, d, 32);
      if (os > bs || (os == bs && oi < bi)) {
        bs = os;
        bi = oi;
      }
    }
    pack |= (unsigned long long)(bi & 31) << (5 * t);
    if (lane == bi) taken = true;
  }

  // gather selected chunks into new flat list; count valid entries
  int vals[NR];
  int cnt = 0;
#pragma unroll
  for (int r = 0; r < NR; ++r) {
    const int f = lane + 32 * r;
    int val = -1;
    if (f < NEWL) {
      const int j = f >> SH;
      const int e = f & (C - 1);
      const int sel = (int)((pack >> (5 * j)) & 31ull);
      const int p = sel * C + e;
      if (LV == 0)
        val = (p <= i) ? p : -1;
      else
        val = s_idx[w][p];
      if (val >= 0) ++cnt;
    }
    vals[r] = val;
  }
#pragma unroll
  for (int d = 16; d > 0; d >>= 1) cnt += __shfl_xor(cnt, d, 32);
  if (lane == 0) s_cnt[w] = cnt;
  __syncthreads();
  int cm = s_cnt[lane];  // global max over all 32 (b,h)
#pragma unroll
  for (int d = 16; d > 0; d >>= 1) {
    const int o = __shfl_xor(cm, d, 32);
    cm = o > cm ? o : cm;
  }
  __syncthreads();
#pragma unroll
  for (int r = 0; r < NR; ++r) {
    const int f = lane + 32 * r;
    if (f < NEWL) {
      const int val = vals[r];
      s_idx[w][f] = (f < cm) ? (val >= 0 ? val : 0) : -1;
    }
  }
  __syncthreads();
  cur_len = cm;
}

__global__ __launch_bounds__(1024) void hip_attn(
    const float* __restrict__ qg, const float* __restrict__ kg,
    const float* __restrict__ vg, float* __restrict__ ctx) {
  __shared__ int s_idx[32][256];
  __shared__ __align__(16) float s_q[32][DH];
  __shared__ int s_cnt[32];
  __shared__ float s_attn[32][16];
  __shared__ int s_g[32][16];

  const int i = blockIdx.x;
  const int tid = threadIdx.x;
  const int w = tid >> 5, lane = tid & 31;
  const int b = w >> 4, h = w & 15;
  const size_t bhBase = ((size_t)b * HH + h) * SS;
  const float* kbh = kg + bhBase * DH;
  const float* vbh = vg + bhBase * DH;

  s_q[w][lane] = qg[(bhBase + i) * DH + lane];
  s_q[w][lane + 32] = qg[(bhBase + i) * DH + lane + 32];
  __syncthreads();

  const float* qlds = &s_q[w][0];
  int cur_len = i + 1;
  prune_level<0>(i, w, lane, cur_len, kbh, qlds, s_idx, s_cnt);
  prune_level<1>(i, w, lane, cur_len, kbh, qlds, s_idx, s_cnt);
  prune_level<2>(i, w, lane, cur_len, kbh, qlds, s_idx, s_cnt);
  prune_level<3>(i, w, lane, cur_len, kbh, qlds, s_idx, s_cnt);
  prune_level<4>(i, w, lane, cur_len, kbh, qlds, s_idx, s_cnt);

  // final sparse attention over 16 kept positions
  const int fidx = (lane < 16) ? s_idx[w][lane] : -1;
  const int g = fidx < 0 ? 0 : fidx;
  float sc = -1e30f;
  if (lane < 16) {
    const float acc = dot64(qlds, kbh + (size_t)g * DH) * 0.125f;  // /sqrt(64)
    sc = (fidx < 0 || g > i) ? -1e30f : acc;
  }
  float m = sc;
#pragma unroll
  for (int d = 16; d > 0; d >>= 1) m = fmaxf(m, __shfl_xor(m, d, 32));
  const float e = expf(sc - m);  // dead / lanes>=16 -> 0
  float ssum = e;
#pragma unroll
  for (int d = 16; d > 0; d >>= 1) ssum += __shfl_xor(ssum, d, 32);
  const float a = e / ssum;
  if (lane < 16) {
    s_attn[w][lane] = a;
    s_g[w][lane] = g;
  }
  __syncthreads();

  float c0 = 0.f, c1 = 0.f;
  const int dh = lane << 1;
#pragma unroll
  for (int l = 0; l < 16; ++l) {
    const float al = s_attn[w][l];
    const float* vr = vbh + (size_t)s_g[w][l] * DH + dh;
    c0 = fmaf(al, vr[0], c0);
    c1 = fmaf(al, vr[1], c1);
  }
  float* crow = ctx + (bhBase + i) * DH;
  crow[dh] = c0;
  crow[dh + 1] = c1;
}

// ---------------------------------------------------------------------------
extern "C" void kernel_launch(void* const* d_in, const int* in_sizes, int n_in,
                              void* d_out, int out_size, void* d_ws,
                              size_t ws_size, hipStream_t stream) {
  (void)in_sizes;
  (void)n_in;
  (void)out_size;
  (void)ws_size;
  const float* x = (const float*)d_in[0];
  const float* Wq = (const float*)d_in[1];
  const float* bq = (const float*)d_in[2];
  const float* Wk = (const float*)d_in[3];
  const float* bk = (const float*)d_in[4];
  const float* Wv = (const float*)d_in[5];
  const float* bv = (const float*)d_in[6];
  const float* Wo = (const float*)d_in[7];
  const float* bo = (const float*)d_in[8];
  float* out = (float*)d_out;

  const size_t elems = (size_t)BB * HH * SS * DH;  // 2M floats
  float* qb = (float*)d_ws;
  float* kb = qb + elems;
  float* vb = kb + elems;
  float* cb = vb + elems;

  dim3 gg(DD / 128, (BB * SS) / 64);
  gemm_wmma_f32<0><<<gg, 256, 0, stream>>>(x, Wq, bq, qb);
  gemm_wmma_f32<0><<<gg, 256, 0, stream>>>(x, Wk, bk, kb);
  gemm_wmma_f32<0><<<gg, 256, 0, stream>>>(x, Wv, bv, vb);
  hip_attn<<<SS, 1024, 0, stream>>>(qb, kb, vb, cb);
  gemm_wmma_f32<1><<<gg, 256, 0, stream>>>(cb, Wo, bo, out);
}